// KDE_lossFun_61624190763244
// MI455X (gfx1250) — compile-verified
//
#include <hip/hip_runtime.h>
#include <math.h>
#include <stdint.h>

typedef float v2f __attribute__((ext_vector_type(2)));
typedef float v8f __attribute__((ext_vector_type(8)));

#define N_ROWS 2048
#define N_CLS  10
#define EPS_F  2.220446049250313e-16f
#define TILE   512
#define WAVES  4

// ws layout (floats): [0]=ce_acc  [1]=ece_acc  [2]=h  [3]=invh2  [4]=scale
//                     [16 .. 16 + N_ROWS*N_CLS) = softmax probs

__global__ void k_init(float* ws) {
    if (threadIdx.x < 16) ws[threadIdx.x] = 0.0f;
}

__global__ void k_softmax_ce(const float* __restrict__ z, const int* __restrict__ lab,
                             float* __restrict__ ws) {
    int row = blockIdx.x * blockDim.x + threadIdx.x;
    if (row >= N_ROWS) return;
    float v[N_CLS];
    float m = -1e30f;
    #pragma unroll
    for (int j = 0; j < N_CLS; ++j) { v[j] = z[row * N_CLS + j]; m = fmaxf(m, v[j]); }
    float s = 0.0f;
    #pragma unroll
    for (int j = 0; j < N_CLS; ++j) { v[j] = expf(v[j] - m); s += v[j]; }
    float inv = 1.0f / s;
    float* probs = ws + 16;
    #pragma unroll
    for (int j = 0; j < N_CLS; ++j) probs[row * N_CLS + j] = v[j] * inv;
    int l = lab[row];
    float logp = (z[row * N_CLS + l] - m) - logf(s);
    atomicAdd(&ws[0], -logp);                      // CE numerator (divide by n at the end)
}

__global__ void k_bandwidth(float* __restrict__ ws) {
    __shared__ float s_sum[N_CLS], s_sq[N_CLS];
    if (threadIdx.x < N_CLS) { s_sum[threadIdx.x] = 0.0f; s_sq[threadIdx.x] = 0.0f; }
    __syncthreads();
    const float* probs = ws + 16;
    float ls[N_CLS] = {0}, lq[N_CLS] = {0};
    for (int r = threadIdx.x; r < N_ROWS; r += blockDim.x) {
        #pragma unroll
        for (int j = 0; j < N_CLS; ++j) {
            float p = probs[r * N_CLS + j];
            ls[j] += p; lq[j] += p * p;
        }
    }
    #pragma unroll
    for (int j = 0; j < N_CLS; ++j) { atomicAdd(&s_sum[j], ls[j]); atomicAdd(&s_sq[j], lq[j]); }
    __syncthreads();
    if (threadIdx.x == 0) {
        float hs = 0.0f;
        #pragma unroll
        for (int j = 0; j < N_CLS; ++j) {
            float var = (s_sq[j] - s_sum[j] * s_sum[j] / (float)N_ROWS) / (float)(N_ROWS - 1);
            hs += 1.06f * sqrtf(fmaxf(var, 0.0f));
        }
        float h = (hs / (float)N_CLS) * powf((float)N_ROWS, -0.2f);
        float base = (35.0f / 32.0f) / h;
        float b2 = base * base, b4 = b2 * b2, b8 = b4 * b4;
        ws[2] = h;
        ws[3] = 1.0f / (h * h);
        ws[4] = b8 * b2;                           // (35/(32h))^10
    }
}

// Issue one async global->LDS 32-bit copy (GVS mode), tracked by ASYNCcnt.
__device__ __forceinline__ void async_g2l_b32(void* lds_dst, const void* g_base,
                                              uint32_t byte_off) {
    uint32_t lds_addr = (uint32_t)(uintptr_t)lds_dst;   // low 32 bits = LDS offset
    asm volatile("global_load_async_to_lds_b32 %0, %1, %2"
                 :
                 : "v"(lds_addr), "v"(byte_off), "s"(g_base)
                 : "memory");
}

__device__ __forceinline__ void wait_async0() {
    asm volatile("s_wait_asynccnt 0" ::: "memory");
}

// 4 waves per block, one block per 16 a-rows. Each wave owns b-chunks
// bb = wv*4 (step 16) and accumulates its own D via V_WMMA_F32_16X16X4_F32:
//   D[16x16] += pi_t[16x4] * B[4x16],
// B = onehot(label[b]) with column 10 forced to 1.0 so D[:,10] == pi_sum.
// Partial Ds are merged through LDS; wave 0 computes the ECE tail.
__global__ __launch_bounds__(128) void k_pairwise_wmma(const int* __restrict__ lab,
                                                       float* __restrict__ ws) {
    __shared__ float sP[TILE * N_CLS];        // 20 KB probs tile
    __shared__ int   sL[TILE];                // 2 KB labels tile
    __shared__ float sAcc[WAVES * 8 * 32];    // 4 KB partial D merge

    const float* probs = ws + 16;
    const float invh2 = ws[3];
    const float scale = ws[4];

    const int tid     = threadIdx.x;
    const int lane    = tid & 31;
    // wv is uniform per wave; readfirstlane makes it an SGPR so the hot b-loop
    // compiles to a uniform scalar-branch loop (no EXEC churn around the WMMA).
    const int wv      = __builtin_amdgcn_readfirstlane(tid >> 5);
    const int rowBase = blockIdx.x * 16;
    const int m       = lane & 15;         // A-matrix row (and D column) this lane holds
    const int half    = lane >> 4;         // 0: K=0,1  1: K=2,3  (ISA 16x4 f32 A layout)
    const int k0      = half * 2;
    const int myRow   = rowBase + m;

    float pa[N_CLS];
    #pragma unroll
    for (int j = 0; j < N_CLS; ++j) pa[j] = probs[myRow * N_CLS + j];

    v8f acc = {};                          // D: lane holds col (lane&15) of rows v + half*8

    for (int t = 0; t < N_ROWS / TILE; ++t) {
        // async-stage tile t into LDS (all 128 threads, exact multiples -> EXEC all 1s)
        const float* gP = probs + t * TILE * N_CLS;
        const int*   gL = lab + t * TILE;
        #pragma unroll 4
        for (int i = tid; i < TILE * N_CLS; i += 128)
            async_g2l_b32(&sP[i], gP, (uint32_t)(i * 4));
        for (int i = tid; i < TILE; i += 128)
            async_g2l_b32(&sL[i], gL, (uint32_t)(i * 4));
        wait_async0();
        __syncthreads();

        for (int bb = wv * 4; bb < TILE; bb += 4 * WAVES) {
            const int bGlob = t * TILE + bb;
            v2f a;
            #pragma unroll
            for (int kk = 0; kk < 2; ++kk) {
                const int bloc = bb + k0 + kk;
                float q = 1.0f;
                #pragma unroll
                for (int j = 0; j < N_CLS; ++j) {
                    float d = pa[j] - sP[bloc * N_CLS + j];
                    q *= fmaxf(1.0f - d * d * invh2, 0.0f);
                }
                float pit = scale * q * q * q;             // relu(t)^3 == relu(t^3)
                if (myRow == bGlob + k0 + kk) pit = 0.0f;  // zero diagonal
                a[kk] = pit;
            }
            v2f b;
            const int l0 = sL[bb + k0], l1 = sL[bb + k0 + 1];
            b[0] = (m == l0 || m == 10) ? 1.0f : 0.0f;
            b[1] = (m == l1 || m == 10) ? 1.0f : 0.0f;
            acc = __builtin_amdgcn_wmma_f32_16x16x4_f32(
                      false, a, false, b, (short)0, acc, false, false);
        }
        __syncthreads();   // everyone done reading sP/sL before next tile overwrites
    }

    // merge the 4 per-wave partial accumulators through LDS
    #pragma unroll
    for (int v = 0; v < 8; ++v) sAcc[(wv * 8 + v) * 32 + lane] = acc[v];
    __syncthreads();

    if (wv == 0) {
        #pragma unroll
        for (int v = 0; v < 8; ++v) {
            acc[v] = sAcc[v * 32 + lane]
                   + sAcc[(8 + v) * 32 + lane]
                   + sAcc[(16 + v) * 32 + lane]
                   + sAcc[(24 + v) * 32 + lane];
        }
        // ECE: per row, sum_c | probs[row,c] - pi[row,c] / (pi_sum[row] + EPS) |
        float local = 0.0f;
        #pragma unroll
        for (int v = 0; v < 8; ++v) {
            const int row     = rowBase + v + half * 8;
            const float pisum = __shfl(acc[v], (lane & 16) + 10, 32);  // col 10 = row sum
            const int c = lane & 15;
            if (c < N_CLS) {
                float pi = acc[v] / (pisum + EPS_F);
                local += fabsf(probs[row * N_CLS + c] - pi);
            }
        }
        #pragma unroll
        for (int off = 16; off > 0; off >>= 1) local += __shfl_xor(local, off, 32);
        if (lane == 0) atomicAdd(&ws[1], local);
    }
}

__global__ void k_final(const float* __restrict__ ws, float* __restrict__ out) {
    if (threadIdx.x == 0) out[0] = (ws[0] + ws[1]) / (float)N_ROWS;
}

extern "C" void kernel_launch(void* const* d_in, const int* in_sizes, int n_in,
                              void* d_out, int out_size, void* d_ws, size_t ws_size,
                              hipStream_t stream) {
    const float* z   = (const float*)d_in[0];
    const int*   lab = (const int*)d_in[1];
    float* ws  = (float*)d_ws;
    float* out = (float*)d_out;

    k_init<<<1, 32, 0, stream>>>(ws);
    k_softmax_ce<<<N_ROWS / 256, 256, 0, stream>>>(z, lab, ws);
    k_bandwidth<<<1, 256, 0, stream>>>(ws);
    k_pairwise_wmma<<<N_ROWS / 16, 128, 0, stream>>>(lab, ws);
    k_final<<<1, 32, 0, stream>>>(ws, out);
}